// PredCodeNet_5042291605831
// MI455X (gfx1250) — compile-verified
//
#include <hip/hip_runtime.h>
#include <hip/hip_bf16.h>

// ---------------------------------------------------------------------------
// PredCodeNet on MI455X (gfx1250, wave32, WMMA).
// Sizes: K=512, N=1024, Z=32, H=256, B=128, T=16.
// All matmuls use V_WMMA_F32_16X16X4_F32 (f32 matrix pipe) — ISTA's
// data-dependent convergence test wants full f32 numerics.
// ---------------------------------------------------------------------------

typedef float v2f __attribute__((ext_vector_type(2)));
typedef float v8f __attribute__((ext_vector_type(8)));

#define PB   128      // batch
#define PT   16       // timesteps
#define PN   1024     // x dim
#define PK   512      // r dim
#define PZ   32       // mixtures
#define PH   256      // rnn hidden
#define LR2  0.1f     // 2 * lr_r
#define LMDA 0.05f
#define TOL2 (0.01f * 0.01f)
#define MAXIT 300

__device__ __forceinline__ v8f wmma4(v2f a, v2f b, v8f c) {
  // D = A(16x4, f32) * B(4x16, f32) + C(16x16, f32)
  return __builtin_amdgcn_wmma_f32_16x16x4_f32(false, a, false, b, (short)0, c,
                                               false, false);
}

// ---------------------------------------------------------------------------
// Generic C = act( A(MxK) * B(NxK)^T + bias + Cin ), one 16x16 tile per wave.
// Branch-free inner loop: OOB B columns are handled by clamping the row
// pointer (their garbage only lands in a C column we never store); the K-tail
// (K % 4 != 0, i.e. K=250) is a single guarded step outside the main loop.
// ---------------------------------------------------------------------------
__global__ void gemm_abT_kernel(const float* __restrict__ A, int lda, long sA,
                                const float* __restrict__ B, int ldb, long sB,
                                const float* __restrict__ bias,
                                const float* __restrict__ Cin,
                                float* __restrict__ C, long sC,
                                int M, int N, int K, int act) {
  const int bz = blockIdx.z;
  A += (long)bz * sA;
  B += (long)bz * sB;
  C += (long)bz * sC;

  const int wave = threadIdx.x >> 5;
  const int lane = threadIdx.x & 31;
  const int half = lane >> 4;   // 0: K=k,k+1 ; 1: K=k+2,k+3
  const int l16  = lane & 15;

  const int Mt = M >> 4;
  const int Nt = (N + 15) >> 4;
  const int tiles = Mt * Nt;
  const int tid = blockIdx.x * 4 + wave;
  if (tid >= tiles) return;                    // uniform per wave

  const int m0 = (tid % Mt) << 4;
  const int n0 = (tid / Mt) << 4;
  const int arow  = m0 + l16;
  const int brow  = n0 + l16;                  // output column index
  const int browc = (brow < N) ? brow : 0;     // clamped: garbage col unsaved

  const float* __restrict__ Ap = A + (long)arow * lda;   // + k
  const float* __restrict__ Bp = B + (long)browc * ldb;  // + k

  const int kmain = K & ~3;
  v8f acc = {};
#pragma unroll 4
  for (int k = 0; k < kmain; k += 4) {
    const int k0 = k + half * 2;
    const v2f a = *(const v2f*)(Ap + k0);      // 8B aligned: lda even, k0 even
    const v2f b = *(const v2f*)(Bp + k0);
    acc = wmma4(a, b, acc);
  }
  if (kmain < K) {                             // single guarded tail (K=250)
    const int k0 = kmain + half * 2;
    v2f a, b;
    a.x = (k0 < K)     ? Ap[k0]     : 0.0f;
    a.y = (k0 + 1 < K) ? Ap[k0 + 1] : 0.0f;
    b.x = (k0 < K)     ? Bp[k0]     : 0.0f;
    b.y = (k0 + 1 < K) ? Bp[k0 + 1] : 0.0f;
    acc = wmma4(a, b, acc);
  }

  const int n = n0 + l16;
  if (n >= N) return;                          // store-side guard only
  const float bv = bias ? bias[n] : 0.0f;
#pragma unroll
  for (int v = 0; v < 8; ++v) {
    const int m = m0 + v + half * 8;
    float cv = acc[v] + bv;
    if (Cin) cv += Cin[(long)m * N + n];
    if (act == 1) cv = tanhf(cv);
    C[(long)m * N + n] = cv;
  }
}

// ---------------------------------------------------------------------------
// Training-mode BatchNorm1d (biased var) + ELU, in place; one column/thread.
// ---------------------------------------------------------------------------
__global__ void bn_elu_kernel(float* __restrict__ z, const float* __restrict__ g,
                              const float* __restrict__ b, int d) {
  const int c = blockIdx.x * blockDim.x + threadIdx.x;
  if (c >= d) return;
  float mu = 0.0f;
  for (int i = 0; i < PB; ++i) mu += z[(long)i * d + c];
  mu *= (1.0f / PB);
  float var = 0.0f;
  for (int i = 0; i < PB; ++i) {
    const float dd = z[(long)i * d + c] - mu;
    var += dd * dd;
  }
  var *= (1.0f / PB);
  const float is = rsqrtf(var + 1e-5f);
  const float gg = g[c], bb = b[c];
  for (int i = 0; i < PB; ++i) {
    const float zn = gg * (z[(long)i * d + c] - mu) * is + bb;
    z[(long)i * d + c] = (zn > 0.0f) ? zn : (expf(zn) - 1.0f);
  }
}

// ---------------------------------------------------------------------------
// r_hat[b,i] = relu( sum_z w[b,z] * Y[z,b,i] )
// ---------------------------------------------------------------------------
__global__ void rhat_kernel(const float* __restrict__ Y,
                            const float* __restrict__ w,
                            float* __restrict__ rhat) {
  const int idx = blockIdx.x * blockDim.x + threadIdx.x;
  if (idx >= PB * PK) return;
  const int b = idx >> 9;      // / PK
  const int i = idx & (PK - 1);
  float s = 0.0f;
#pragma unroll
  for (int z = 0; z < PZ; ++z)
    s += w[b * PZ + z] * Y[((long)z * PB + b) * PK + i];
  rhat[idx] = fmaxf(s, 0.0f);
}

// ---------------------------------------------------------------------------
// Persistent ISTA: 8 blocks x 256 threads (8 waves). Block owns 16 batch rows.
// LDS: rcur 16x512 (32KB) + err 16x1024 (64KB) + reduction scratch.
// Branch-free inner loops, float2 fragment loads (ds_load_b64 from LDS,
// global_load_b64 for W_U rows).
// ---------------------------------------------------------------------------
__global__ void ista_kernel(const float* __restrict__ x, int t,
                            const float* __restrict__ W_U,
                            const float* __restrict__ rhat,
                            float* __restrict__ r_out,
                            float* __restrict__ ilpart,
                            float* __restrict__ tlpart) {
  extern __shared__ float smem[];
  float* rcur = smem;                 // 16*512
  float* err  = smem + 16 * PK;       // 16*1024
  float* red  = err + 16 * PN;        // [0]=diff [1]=nrm [2]=flag

  const int m0   = blockIdx.x * 16;
  const int wave = threadIdx.x >> 5;
  const int lane = threadIdx.x & 31;
  const int half = lane >> 4;
  const int l16  = lane & 15;
  const int koff = half * 2;

  for (int i = threadIdx.x; i < 16 * PK; i += blockDim.x)
    rcur[i] = rhat[(long)m0 * PK + i];
  if (threadIdx.x == 0) { red[0] = 0.0f; red[1] = 0.0f; }
  __syncthreads();

  for (int it = 0; it < MAXIT; ++it) {
    // ---- phase 1: err = rcur * W_U^T - x_t ----
    for (int tt = 0; tt < 8; ++tt) {
      const int n0 = (wave * 8 + tt) << 4;
      const float* __restrict__ Wp = W_U + (long)(n0 + l16) * PK;
      const float* __restrict__ Rp = rcur + l16 * PK;
      v8f acc = {};
#pragma unroll 4
      for (int k = 0; k < PK; k += 4) {
        const v2f a = *(const v2f*)(Rp + k + koff);
        const v2f b = *(const v2f*)(Wp + k + koff);
        acc = wmma4(a, b, acc);
      }
      const int n = n0 + l16;
#pragma unroll
      for (int v = 0; v < 8; ++v) {
        const int mr = v + half * 8;
        err[mr * PN + n] =
            acc[v] - x[((long)(m0 + mr) * PT + t) * PN + n];
      }
    }
    __syncthreads();   // err visible; red[0..1] zeroed

    // ---- phase 2: g = err * W_U ; soft-threshold update ----
    float ldiff = 0.0f, lnrm = 0.0f;
    for (int tt = 0; tt < 4; ++tt) {
      const int c0 = (wave * 4 + tt) << 4;
      const float* __restrict__ Ep = err + l16 * PN;
      const float* __restrict__ Wp = W_U + c0 + l16;
      v8f acc = {};
#pragma unroll 2
      for (int nb = 0; nb < PN; nb += 4) {
        const int n0i = nb + koff;
        const v2f a = *(const v2f*)(Ep + n0i);
        v2f b;
        b.x = Wp[(long)n0i * PK];
        b.y = Wp[(long)(n0i + 1) * PK];
        acc = wmma4(a, b, acc);
      }
#pragma unroll
      for (int v = 0; v < 8; ++v) {
        const int mr  = v + half * 8;
        const int idx = mr * PK + c0 + l16;
        const float old = rcur[idx];
        float rn = old - LR2 * acc[v] - LMDA;   // r - lr*2*(err@W_U) - lmda
        rn = (rn > 0.0f) ? rn : 0.0f;           // soft-threshold (nonneg)
        const float dd = rn - old;
        ldiff += dd * dd;
        lnrm  += old * old;
        rcur[idx] = rn;
      }
    }
    atomicAdd(&red[0], ldiff);
    atomicAdd(&red[1], lnrm);
    __syncthreads();
    if (threadIdx.x == 0) {
      red[2] = (red[0] < TOL2 * red[1]) ? 1.0f : 0.0f;
      red[0] = 0.0f;
      red[1] = 0.0f;
    }
    __syncthreads();
    if (red[2] != 0.0f) break;
  }

  // ---- losses with final rcur ----
  float lil = 0.0f;
  for (int tt = 0; tt < 8; ++tt) {
    const int n0 = (wave * 8 + tt) << 4;
    const float* __restrict__ Wp = W_U + (long)(n0 + l16) * PK;
    const float* __restrict__ Rp = rcur + l16 * PK;
    v8f acc = {};
#pragma unroll 4
    for (int k = 0; k < PK; k += 4) {
      const v2f a = *(const v2f*)(Rp + k + koff);
      const v2f b = *(const v2f*)(Wp + k + koff);
      acc = wmma4(a, b, acc);
    }
    const int n = n0 + l16;
#pragma unroll
    for (int v = 0; v < 8; ++v) {
      const int mr = v + half * 8;
      const float d = acc[v] - x[((long)(m0 + mr) * PT + t) * PN + n];
      lil += d * d;
    }
  }
  float ltl = 0.0f;
  for (int i = threadIdx.x; i < 16 * PK; i += blockDim.x) {
    const float rv = rcur[i];
    const float dv = rv - rhat[(long)m0 * PK + i];
    ltl += dv * dv;
    r_out[(long)m0 * PK + i] = rv;
  }
  atomicAdd(&red[0], lil);
  atomicAdd(&red[1], ltl);
  __syncthreads();
  if (threadIdx.x == 0) {
    ilpart[blockIdx.x] = red[0];
    tlpart[blockIdx.x] = red[1];
  }
}

// Deterministic fixed-order accumulation of per-block partials.
__global__ void loss_accum_kernel(const float* __restrict__ ilp,
                                  const float* __restrict__ tlp,
                                  float* __restrict__ out) {
  if (threadIdx.x == 0 && blockIdx.x == 0) {
    float a = 0.0f, b = 0.0f;
    for (int i = 0; i < 8; ++i) { a += ilp[i]; b += tlp[i]; }
    out[0] += a;
    out[1] += b;
  }
}

__global__ void zero_kernel(float* __restrict__ p, int n) {
  const int i = blockIdx.x * blockDim.x + threadIdx.x;
  if (i < n) p[i] = 0.0f;
}

// ---------------------------------------------------------------------------
// Host orchestration (graph-capture safe: fixed launch sequence, stream-only).
// ---------------------------------------------------------------------------
static inline void launch_gemm(const float* A, int lda, long sA,
                               const float* B, int ldb, long sB,
                               const float* bias, const float* Cin,
                               float* C, long sC,
                               int M, int N, int K, int act, int batch,
                               hipStream_t s) {
  const int Mt = M >> 4;
  const int Nt = (N + 15) >> 4;
  const int tiles = Mt * Nt;
  dim3 grid((tiles + 3) / 4, 1, batch);
  gemm_abT_kernel<<<grid, 128, 0, s>>>(A, lda, sA, B, ldb, sB, bias, Cin, C,
                                       sC, M, N, K, act);
}

extern "C" void kernel_launch(void* const* d_in, const int* in_sizes, int n_in,
                              void* d_out, int out_size, void* d_ws,
                              size_t ws_size, hipStream_t stream) {
  const float* x    = (const float*)d_in[0];   // (128,16,1024)
  const float* W_U  = (const float*)d_in[1];   // (1024,512)
  const float* V    = (const float*)d_in[2];   // (32,512,512)
  const float* W_ih = (const float*)d_in[3];   // (256,512)
  const float* W_hh = (const float*)d_in[4];   // (256,256)
  const float* b_ih = (const float*)d_in[5];
  const float* b_hh = (const float*)d_in[6];
  const float* fcw[6];
  const float* fcb[6];
  const float* bng[5];
  const float* bnb[5];
  for (int i = 0; i < 6; ++i) fcw[i] = (const float*)d_in[7 + i];
  for (int i = 0; i < 6; ++i) fcb[i] = (const float*)d_in[13 + i];
  for (int i = 0; i < 5; ++i) bng[i] = (const float*)d_in[19 + i];
  for (int i = 0; i < 5; ++i) bnb[i] = (const float*)d_in[24 + i];
  float* out = (float*)d_out;

  // workspace layout (floats)
  float* ws   = (float*)d_ws;
  float* r    = ws;                 // 128*512      = 65536
  float* hA   = r + 65536;          // 128*256      = 32768
  float* hB   = hA + 32768;         // 32768
  float* z0   = hB + 32768;         // 65536 (layers up to 500 wide)
  float* z1   = z0 + 65536;         // 65536
  float* wbuf = z1 + 65536;         // 128*32       = 4096
  float* rhat = wbuf + 4096;        // 65536
  float* ilp  = rhat + 65536;       // 8
  float* tlp  = ilp + 16;           // 8 (padded)
  float* Y    = tlp + 48;           // 32*128*512   = 2097152

  const int dims[7] = {PH, 500, 250, 250, 100, 100, PZ};

  // zero-init state (harness poisons buffers; we must not rely on them)
  zero_kernel<<<(65536 + 255) / 256, 256, 0, stream>>>(r, 65536);
  zero_kernel<<<(32768 + 255) / 256, 256, 0, stream>>>(hA, 32768);
  zero_kernel<<<1, 32, 0, stream>>>(out, 2);

  const size_t ista_shmem = (size_t)(16 * PK + 16 * PN + 16) * sizeof(float);

  for (int t = 0; t < PT; ++t) {
    float* hprev = (t & 1) ? hB : hA;
    float* hnext = (t & 1) ? hA : hB;

    // RNN cell: h' = tanh(r @ W_ih^T + b_ih + h @ W_hh^T + b_hh)
    launch_gemm(r, PK, 0, W_ih, PK, 0, b_ih, nullptr, z0, 0,
                PB, PH, PK, 0, 1, stream);
    launch_gemm(hprev, PH, 0, W_hh, PH, 0, b_hh, z0, hnext, 0,
                PB, PH, PH, 1, 1, stream);

    // BN-ELU MLP
    const float* in = hnext;
    int din = PH;
    for (int l = 0; l < 5; ++l) {
      float* zb = (l & 1) ? z1 : z0;
      const int dout = dims[l + 1];
      launch_gemm(in, din, 0, fcw[l], din, 0, fcb[l], nullptr, zb, 0,
                  PB, dout, din, 0, 1, stream);
      bn_elu_kernel<<<(dout + 255) / 256, 256, 0, stream>>>(zb, bng[l],
                                                            bnb[l], dout);
      in = zb;
      din = dout;
    }
    // w = z @ fcw[5]^T + fcb[5]   -> (128, 32)
    launch_gemm(in, din, 0, fcw[5], din, 0, fcb[5], nullptr, wbuf, 0,
                PB, PZ, din, 0, 1, stream);

    // Y[z] = r @ V[z]^T  (32-way batched 128x512x512 GEMM)
    launch_gemm(r, PK, 0, V, PK, (long)PK * PK, nullptr, nullptr, Y,
                (long)PB * PK, PB, PK, PK, 0, PZ, stream);

    // r_hat = relu(sum_z w_z * Y[z])
    rhat_kernel<<<(PB * PK + 255) / 256, 256, 0, stream>>>(Y, wbuf, rhat);

    // ISTA (persistent, in-kernel convergence loop) + loss partials
    ista_kernel<<<8, 256, ista_shmem, stream>>>(x, t, W_U, rhat, r, ilp, tlp);
    loss_accum_kernel<<<1, 32, 0, stream>>>(ilp, tlp, out);
  }
}